// NodeModel_62921270886987
// MI455X (gfx1250) — compile-verified
//
#include <hip/hip_runtime.h>
#include <hip/hip_bf16.h>

typedef __attribute__((ext_vector_type(2))) float v2f;
typedef __attribute__((ext_vector_type(4))) float v4f;
typedef __attribute__((ext_vector_type(8))) float v8f;

#define FX    128
#define FE    128
#define DIN   256
#define HID   256
#define FOUT  128
#define MTILE 48    // nodes per block (3 x 16 M-subtiles)
#define LDST  260   // LDS row stride (floats): 260 % 64 = 4 -> conflict-free column reads

// ---------------- Kernel 1: scatter-add of edge features + counts ----------------
// One 32-lane group per edge. Non-temporal 16B loads keep the one-shot 410 MB edge
// stream from evicting the L2-resident sums accumulator; f32 atomics resolve in L2.
__global__ __launch_bounds__(256)
void scatter_edges_kernel(const float* __restrict__ edge_attr,
                          const long long* __restrict__ dst,
                          float* __restrict__ sums,
                          float* __restrict__ counts,
                          int n_edges) {
    int e = blockIdx.x * 8 + (threadIdx.x >> 5);
    if (e >= n_edges) return;
    int lane = threadIdx.x & 31;
    long long d = dst[e];
    const v4f* ea = reinterpret_cast<const v4f*>(edge_attr + (size_t)e * FE);
    v4f v = __builtin_nontemporal_load(ea + lane);   // global_load_b128, TH=NT
    float* s = sums + (size_t)d * FE + lane * 4;
    atomicAdd(s + 0, v.x);
    atomicAdd(s + 1, v.y);
    atomicAdd(s + 2, v.z);
    atomicAdd(s + 3, v.w);
    if (lane == 0) atomicAdd(counts + d, 1.0f);
}

// ---------------- Kernel 2: fused mean + concat + MLP (WMMA f32) ----------------
// Block = 256 threads = 8 wave32, handles MTILE=48 nodes.
// GEMM1: h[48,256] = relu(NI[48,256] @ W1[256,256] + b1); wave w owns cols [32w,32w+32),
//        3 M-subtiles share each B fragment (0.67 B-loads per WMMA).
// GEMM2: out[48,128] = h @ W2[256,128] + b2; wave w owns cols [16w,16w+16).
// Single LDS tile buffer: NI during GEMM1 (acc in VGPRs), overwritten by h for GEMM2.
__global__ __launch_bounds__(256)
void node_mlp_kernel(const float* __restrict__ x,
                     const float* __restrict__ sums,
                     const float* __restrict__ counts,
                     const float* __restrict__ W1,
                     const float* __restrict__ b1,
                     const float* __restrict__ W2,
                     const float* __restrict__ b2,
                     float* __restrict__ out,
                     int n_nodes) {
    __shared__ float s_tile[MTILE * LDST];   // NI tile, then reused as h tile
    __shared__ float s_inv[MTILE];

    const int t = threadIdx.x;
    const int block_row = blockIdx.x * MTILE;

    if (t < MTILE) {
        int node = block_row + t;
        float c = (node < n_nodes) ? counts[node] : 1.0f;
        s_inv[t] = 1.0f / fmaxf(c, 1.0f);
    }
    __syncthreads();

    // Build node_inputs tile: cols [0,128)=x, [128,256)=mean-aggregated edge attrs.
    // float4 granularity: 48*64 = 3072 vectors, 12 per thread.
    for (int idx = t; idx < MTILE * (DIN / 4); idx += 256) {
        int row = idx >> 6;            // DIN/4 = 64
        int c4  = (idx & 63) * 4;
        int node = block_row + row;
        v4f v = {0.0f, 0.0f, 0.0f, 0.0f};
        if (node < n_nodes) {
            if (c4 < FX) {
                v = *reinterpret_cast<const v4f*>(x + (size_t)node * FX + c4);
            } else {
                v4f sv = *reinterpret_cast<const v4f*>(sums + (size_t)node * FE + (c4 - FX));
                v = sv * s_inv[row];
            }
        }
        *reinterpret_cast<v4f*>(&s_tile[row * LDST + c4]) = v;   // 1040B row stride: 16B aligned
    }
    __syncthreads();

    const int lane = t & 31;
    const int wave = t >> 5;
    const int half = lane >> 4;   // 0: lanes 0-15 hold K,K+1 ; 1: lanes 16-31 hold K+2,K+3
    const int l16  = lane & 15;

    // ---------------- GEMM1 (accumulators stay in VGPRs) ----------------
    const int c0 = wave * 32;
    const int c1 = c0 + 16;
    v8f acc1v[3][2] = {};
    for (int k = 0; k < DIN; k += 4) {
        const int kk = k + 2 * half;
        const float* w1k0 = W1 + (size_t)kk * HID;
        const float* w1k1 = w1k0 + HID;
        v2f b0, b1v;
        b0.x  = w1k0[c0 + l16]; b0.y  = w1k1[c0 + l16];
        b1v.x = w1k0[c1 + l16]; b1v.y = w1k1[c1 + l16];
        #pragma unroll
        for (int m = 0; m < 3; ++m) {
            const float* ar = &s_tile[(m * 16 + l16) * LDST + kk];
            v2f a;
            a.x = ar[0];
            a.y = ar[1];
            acc1v[m][0] = __builtin_amdgcn_wmma_f32_16x16x4_f32(false, a, false, b0,  (short)0, acc1v[m][0], false, false);
            acc1v[m][1] = __builtin_amdgcn_wmma_f32_16x16x4_f32(false, a, false, b1v, (short)0, acc1v[m][1], false, false);
        }
    }
    const float bias0 = b1[c0 + l16];
    const float bias1 = b1[c1 + l16];
    __syncthreads();   // everyone done reading NI; tile becomes h

    // C/D layout: VGPR i -> row i (lanes 0-15) / i+8 (lanes 16-31), col = l16
    #pragma unroll
    for (int m = 0; m < 3; ++m) {
        #pragma unroll
        for (int i = 0; i < 8; ++i) {
            int row = m * 16 + i + 8 * half;
            s_tile[row * LDST + c0 + l16] = fmaxf(acc1v[m][0][i] + bias0, 0.0f);
            s_tile[row * LDST + c1 + l16] = fmaxf(acc1v[m][1][i] + bias1, 0.0f);
        }
    }
    __syncthreads();

    // ---------------- GEMM2 ----------------
    const int oc = wave * 16;
    v8f acc2[3] = {};
    for (int k = 0; k < HID; k += 4) {
        const int kk = k + 2 * half;
        v2f b;
        b.x = W2[(size_t)kk * FOUT + oc + l16];
        b.y = W2[(size_t)(kk + 1) * FOUT + oc + l16];
        #pragma unroll
        for (int m = 0; m < 3; ++m) {
            const float* ar = &s_tile[(m * 16 + l16) * LDST + kk];
            v2f a;
            a.x = ar[0];
            a.y = ar[1];
            acc2[m] = __builtin_amdgcn_wmma_f32_16x16x4_f32(false, a, false, b, (short)0, acc2[m], false, false);
        }
    }
    const float bias = b2[oc + l16];
    #pragma unroll
    for (int m = 0; m < 3; ++m) {
        #pragma unroll
        for (int i = 0; i < 8; ++i) {
            int row = m * 16 + i + 8 * half;
            int node = block_row + row;
            if (node < n_nodes)
                __builtin_nontemporal_store(acc2[m][i] + bias,
                                            &out[(size_t)node * FOUT + oc + l16]);
        }
    }
}

extern "C" void kernel_launch(void* const* d_in, const int* in_sizes, int n_in,
                              void* d_out, int out_size, void* d_ws, size_t ws_size,
                              hipStream_t stream) {
    const float*     x          = (const float*)d_in[0];
    const long long* edge_index = (const long long*)d_in[1];
    const float*     edge_attr  = (const float*)d_in[2];
    const float*     W1         = (const float*)d_in[3];
    const float*     b1         = (const float*)d_in[4];
    const float*     W2         = (const float*)d_in[5];
    const float*     b2         = (const float*)d_in[6];
    float*           out        = (float*)d_out;

    const int n_nodes = in_sizes[0] / FX;
    const int n_edges = in_sizes[1] / 2;
    const long long* dst = edge_index + n_edges;   // edge_index[1, :]

    // Workspace: sums [n_nodes, FE] f32, counts [n_nodes] f32 (~25.8 MB, L2-resident)
    float* sums = (float*)d_ws;
    float* cnts = sums + (size_t)n_nodes * FE;
    size_t ws_floats = (size_t)n_nodes * FE + (size_t)n_nodes;
    hipMemsetAsync(d_ws, 0, ws_floats * sizeof(float), stream);

    const int edge_blocks = (n_edges + 7) / 8;          // 8 edges per 256-thread block
    scatter_edges_kernel<<<edge_blocks, 256, 0, stream>>>(edge_attr, dst, sums, cnts, n_edges);

    const int node_blocks = (n_nodes + MTILE - 1) / MTILE;
    node_mlp_kernel<<<node_blocks, 256, 0, stream>>>(x, sums, cnts, W1, b1, W2, b2, out, n_nodes);
}